// SinimLossOld_63720134803981
// MI455X (gfx1250) — compile-verified
//
#include <hip/hip_runtime.h>

#define BLOCK 256
#define ROWS_PER_THREAD 4                      // 4 rows * 5 floats = 20 floats = 5 float4 per thread per stage
#define F4_PER_THREAD 5
#define CHUNK_ROWS (BLOCK * ROWS_PER_THREAD)   // 1024 rows per block-stage
#define GRID 1024

// ---- CDNA5 async global->LDS (ASYNCcnt path) --------------------------------
// ISA 15.18.3 opcode 98: GLOBAL_LOAD_ASYNC_TO_LDS_B128. Per-lane copy of 16B
// from global addr (VADDR pair) to LDS byte address (VDST). INST_OFFSET is
// added to BOTH the LDS and the global address, so one base each + offset:imm
// covers all 5 float4s of a thread's stage region.
__device__ __forceinline__ void async_ld_b128(unsigned lds_byte_addr, const float4* gsrc, int imm_off) {
    asm volatile("global_load_async_to_lds_b128 %0, %1, off offset:%c2"
                 :: "v"(lds_byte_addr), "v"(gsrc), "i"(imm_off)
                 : "memory");
}

#define ASYNC_WAIT(N) asm volatile("s_wait_asynccnt %c0" :: "i"(N) : "memory")
// Guarantee all prior DS reads of the buffer being recycled have completed
// before the async engine is allowed to overwrite it (WAR on LDS).
#define DS_DRAIN()    asm volatile("s_wait_dscnt 0x0" ::: "memory")

// ---- per-element weighted square -------------------------------------------
// w[i][j] = (i==j) ? 0 : 2*(1+|i-j|)  ==>  term = 4 * (o*u)^2, u = ad + min(ad,1)
__device__ __forceinline__ float wacc(float o, float lf, float cf, float acc) {
    float ad = __builtin_fabsf(lf - cf);          // |label - col| (abs = free src mod)
    float u  = ad + __builtin_fminf(ad, 1.0f);    // {0,2,3,4,5}
    float p  = o * u;
    return __builtin_fmaf(p, p, acc);
}

__device__ __forceinline__ void issue_stage(const float4* g4, long long rowBase,
                                            int tid, const float4* sreg) {
    const float4* g = g4 + (rowBase * 5LL) / 4LL + (long long)tid * F4_PER_THREAD;
    unsigned lds = (unsigned)(unsigned long long)(sreg + tid * F4_PER_THREAD);
    async_ld_b128(lds, g, 0);
    async_ld_b128(lds, g, 16);
    async_ld_b128(lds, g, 32);
    async_ld_b128(lds, g, 48);
    async_ld_b128(lds, g, 64);
}

// 4 independent accumulators (one per row owned by this thread) -> 4x FMA ILP.
__device__ __forceinline__ void stage_compute(const float4* sreg, const int* __restrict__ labels,
                                              long long rowBase, int tid, float acc[4]) {
    const long long r0 = rowBase + (long long)tid * ROWS_PER_THREAD;
    const int4 lb = *(const int4*)(labels + r0);                 // coalesced 16B
    const float l0 = (float)lb.x, l1 = (float)lb.y, l2 = (float)lb.z, l3 = (float)lb.w;
    const float4* s = sreg + tid * F4_PER_THREAD;
    float4 a = s[0], b = s[1], c = s[2], d = s[3], e = s[4];     // ds_load_b128 x5
    // flat layout: row0 = a.xyzw b.x | row1 = b.yzw c.xy | row2 = c.zw d.xyz | row3 = d.w e.xyzw
    acc[0] = wacc(a.x, l0, 0.f, acc[0]); acc[0] = wacc(a.y, l0, 1.f, acc[0]);
    acc[0] = wacc(a.z, l0, 2.f, acc[0]); acc[0] = wacc(a.w, l0, 3.f, acc[0]);
    acc[0] = wacc(b.x, l0, 4.f, acc[0]);
    acc[1] = wacc(b.y, l1, 0.f, acc[1]); acc[1] = wacc(b.z, l1, 1.f, acc[1]);
    acc[1] = wacc(b.w, l1, 2.f, acc[1]); acc[1] = wacc(c.x, l1, 3.f, acc[1]);
    acc[1] = wacc(c.y, l1, 4.f, acc[1]);
    acc[2] = wacc(c.z, l2, 0.f, acc[2]); acc[2] = wacc(c.w, l2, 1.f, acc[2]);
    acc[2] = wacc(d.x, l2, 2.f, acc[2]); acc[2] = wacc(d.y, l2, 3.f, acc[2]);
    acc[2] = wacc(d.z, l2, 4.f, acc[2]);
    acc[3] = wacc(d.w, l3, 0.f, acc[3]); acc[3] = wacc(e.x, l3, 1.f, acc[3]);
    acc[3] = wacc(e.y, l3, 2.f, acc[3]); acc[3] = wacc(e.z, l3, 3.f, acc[3]);
    acc[3] = wacc(e.w, l3, 4.f, acc[3]);
}

__global__ __launch_bounds__(BLOCK) void sinim_partial_kernel(const float* __restrict__ outputs,
                                                              const int* __restrict__ labels,
                                                              float* __restrict__ partials,
                                                              long long B) {
    __shared__ float4 sbuf[2][BLOCK * F4_PER_THREAD];   // 40 KB double buffer
    __shared__ float  red[BLOCK];
    const int tid = threadIdx.x;
    const float4* g4 = (const float4*)outputs;

    const long long stride = (long long)gridDim.x * CHUNK_ROWS;
    long long rb = (long long)blockIdx.x * CHUNK_ROWS;
    float acc[4] = {0.0f, 0.0f, 0.0f, 0.0f};

    // ---- double-buffered async pipeline over full 1024-row stages ----
    bool has = (rb + CHUNK_ROWS <= B);
    int cur = 0;
    if (has) issue_stage(g4, rb, tid, &sbuf[0][0]);
    while (has) {
        long long nx = rb + stride;
        bool hasN = (nx + CHUNK_ROWS <= B);
        if (hasN) {
            DS_DRAIN();                  // buffer cur^1 was read last iter; reads must land first
            issue_stage(g4, nx, tid, &sbuf[cur ^ 1][0]);
            ASYNC_WAIT(F4_PER_THREAD);   // oldest 5 (current stage) complete, next 5 in flight
        } else {
            ASYNC_WAIT(0);
        }
        stage_compute(&sbuf[cur][0], labels, rb, tid, acc);
        rb = nx; has = hasN; cur ^= 1;
    }

    float accs = (acc[0] + acc[1]) + (acc[2] + acc[3]);

    // ---- partial tail stage (rows [B/1024*1024, B)), direct loads ----
    long long tailBase = (B / CHUNK_ROWS) * CHUNK_ROWS;
    if (tailBase < B) {
        int owner = (int)((tailBase / CHUNK_ROWS) % (long long)gridDim.x);
        if ((int)blockIdx.x == owner) {
            for (long long r = tailBase + tid; r < B; r += BLOCK) {
                const float* rp = outputs + r * 5LL;
                float lf = (float)labels[r];
                accs = wacc(rp[0], lf, 0.f, accs);
                accs = wacc(rp[1], lf, 1.f, accs);
                accs = wacc(rp[2], lf, 2.f, accs);
                accs = wacc(rp[3], lf, 3.f, accs);
                accs = wacc(rp[4], lf, 4.f, accs);
            }
        }
    }

    // ---- deterministic block reduction ----
    red[tid] = accs;
    __syncthreads();
    for (int s = BLOCK / 2; s > 0; s >>= 1) {
        if (tid < s) red[tid] += red[tid + s];
        __syncthreads();
    }
    if (tid == 0) partials[blockIdx.x] = red[0];
}

__global__ __launch_bounds__(BLOCK) void sinim_final_kernel(const float* __restrict__ partials,
                                                            int n, float* __restrict__ out,
                                                            float scale) {
    __shared__ float red[BLOCK];
    float a = 0.0f;
    for (int i = threadIdx.x; i < n; i += BLOCK) a += partials[i];
    red[threadIdx.x] = a;
    __syncthreads();
    for (int s = BLOCK / 2; s > 0; s >>= 1) {
        if ((int)threadIdx.x < s) red[threadIdx.x] += red[threadIdx.x + s];
        __syncthreads();
    }
    if (threadIdx.x == 0) out[0] = red[0] * scale;   // scale = 4 / B
}

extern "C" void kernel_launch(void* const* d_in, const int* in_sizes, int n_in,
                              void* d_out, int out_size, void* d_ws, size_t ws_size,
                              hipStream_t stream) {
    (void)n_in; (void)out_size;
    const float* outputs = (const float*)d_in[0];
    const int*   labels  = (const int*)d_in[1];
    long long B = (long long)in_sizes[1];          // labels count; outputs = 5*B floats
    float* partials = (float*)d_ws;

    int grid = GRID;
    if ((size_t)grid * sizeof(float) > ws_size) grid = (int)(ws_size / sizeof(float));
    if (grid < 1) grid = 1;

    sinim_partial_kernel<<<grid, BLOCK, 0, stream>>>(outputs, labels, partials, B);
    sinim_final_kernel<<<1, BLOCK, 0, stream>>>(partials, grid, (float*)d_out,
                                                (float)(4.0 / (double)B));
}